// SelectiveSSMKernel_33002528702788
// MI455X (gfx1250) — compile-verified
//
#include <hip/hip_runtime.h>
#include <hip/hip_bf16.h>

// ---------------------------------------------------------------------------
// Selective SSM forward for MI455X (gfx1250), exploiting the algebraic
// collapse of the reference: h is d-independent -> y is rank-1 in d.
//
// Pipeline (all on `stream`):
//   1) k_prep_weights : M_B = W_B @ W_in (16x1024), pack [M_B ; W_C] into a
//                       K-pair-interleaved layout for b64 WMMA B-operand loads
//   2) k_wsum         : w_sum[e] = sum_d W_out[e,d] ; c2[e] = w_sum*norm_w
//   3) k_msq          : msq = mean(w_sum^2)
//   4) k_gemm_bc      : [Bx|Cx] = u_bld @ Wt^T via V_WMMA_F32_16X16X4_F32,
//                       K split 4 ways for latency hiding (partials summed later)
//   5) k_scan_local   : 32-step local scans of h (in place), chunk finals out
//   6) k_scan_carry   : carry scan across 64 chunks with decay^32
//   7) k_combine      : H = local + decay^(i+1)*carry ; s = sum_n Cx*H ;
//                       g = s * rsqrt(s^2*msq + eps)     (float4 loads)
//   8) k_outer        : out[b,e,l] = g[b,l] * c2[e]      (rank-1, float4 stores)
// ---------------------------------------------------------------------------

typedef __attribute__((ext_vector_type(2))) float v2f;
typedef __attribute__((ext_vector_type(8))) float v8f;

constexpr int kB = 4;
constexpr int kD = 1024;   // D == d_inner
constexpr int kL = 2048;
constexpr int kN = 16;
constexpr int kSplit = 4;          // K-split for the WMMA GEMM
constexpr int kKS = kD / kSplit;   // 256 d-values per partial
constexpr int kChunk = 32;
constexpr int kNChunk = kL / kChunk;   // 64
constexpr float kAScale = 0.1f;
constexpr float kEps = 1.1920928955078125e-07f;  // finfo(float32).eps

// workspace layout (float offsets)
constexpr size_t BCP_SZ   = (size_t)kB * kL * 32;             // one partial [b][l][32]
constexpr size_t WS_WT    = 0;                                // 32*kD
constexpr size_t WS_BCP   = WS_WT + (size_t)32 * kD;
constexpr size_t WS_CARRY = WS_BCP + (size_t)kSplit * BCP_SZ;
constexpr size_t WS_G     = WS_CARRY + (size_t)kB * kN * kNChunk;
constexpr size_t WS_WSUM  = WS_G + (size_t)kB * kL;
constexpr size_t WS_C2    = WS_WSUM + kD;
constexpr size_t WS_MSQ   = WS_C2 + kD;

// ---------------------------------------------------------------------------
// 1) Packed weights. For each d-pair p = d>>1 (64-float block):
//      WtP[p*64 +      n*2 + (d&1)] = M_B[n][d]   (Bx weights, n = 0..15)
//      WtP[p*64 + 32 + n*2 + (d&1)] = W_C[n][d]   (Cx weights)
//    so a lane's WMMA B-operand (K-rows d0, d0+1 at column n) is one b64 load.
__global__ void k_prep_weights(const float* __restrict__ W_in,
                               const float* __restrict__ W_B,
                               const float* __restrict__ W_C,
                               float* __restrict__ WtP) {
  int d = blockIdx.x * blockDim.x + threadIdx.x;  // 0..1023
  float acc[kN];
#pragma unroll
  for (int n = 0; n < kN; ++n) acc[n] = 0.f;
  for (int e = 0; e < kD; ++e) {
    float win = W_in[(size_t)e * kD + d];          // coalesced over d
#pragma unroll
    for (int n = 0; n < kN; ++n)
      acc[n] = fmaf(W_B[(size_t)n * kD + e], win, acc[n]);  // uniform -> scalar path
  }
  size_t blk = (size_t)(d >> 1) * 64 + (d & 1);
#pragma unroll
  for (int n = 0; n < kN; ++n) {
    WtP[blk + n * 2]      = acc[n];
    WtP[blk + 32 + n * 2] = W_C[(size_t)n * kD + d];
  }
}

// ---------------------------------------------------------------------------
// 2) row sums of W_out; one wave per row e
__global__ void k_wsum(const float* __restrict__ W_out,
                       const float* __restrict__ norm_w,
                       float* __restrict__ w_sum, float* __restrict__ c2) {
  int wave = threadIdx.x >> 5, lane = threadIdx.x & 31;
  int e = blockIdx.x * 8 + wave;
  float s = 0.f;
  for (int i = lane; i < kD; i += 32) s += W_out[(size_t)e * kD + i];
#pragma unroll
  for (int m = 16; m; m >>= 1) s += __shfl_xor(s, m, 32);
  if (lane == 0) { w_sum[e] = s; c2[e] = s * norm_w[e]; }
}

// 3) msq = mean(w_sum^2)
__global__ void k_msq(const float* __restrict__ w_sum, float* __restrict__ msq) {
  __shared__ float sm[256];
  float s = 0.f;
  for (int i = threadIdx.x; i < kD; i += 256) { float v = w_sum[i]; s += v * v; }
  sm[threadIdx.x] = s;
  __syncthreads();
  for (int off = 128; off; off >>= 1) {
    if ((int)threadIdx.x < off) sm[threadIdx.x] += sm[threadIdx.x + off];
    __syncthreads();
  }
  if (threadIdx.x == 0) msq[0] = sm[0] * (1.f / (float)kD);
}

// ---------------------------------------------------------------------------
// 4) [Bx|Cx](b,l, 0:32) = sum_d u[b,d,l] * W[d][0:32]  via f32 WMMA.
//    Per wave: one 16(l) x 16(n) tile for Bx and one for Cx, K-split 4 ways.
//    A (16x4 f32, ISA layout): lanes 0-15 M=lane K={0,1}; lanes 16-31 M=lane-16 K={2,3}.
//    B (4x16): VGPR0 = rows K=0 (lanes 0-15) / K=2 (lanes 16-31), VGPR1 = K=1/K=3;
//              packed layout makes each lane's (K,K+1) pair one b64 load.
//    D (16x16 f32): VGPR r -> M=r (lanes 0-15) / M=r+8 (lanes 16-31), N=lane&15.
__global__ void k_gemm_bc(const float* __restrict__ u,
                          const float* __restrict__ WtP,
                          float* __restrict__ BCp) {
  int wave = threadIdx.x >> 5, lane = threadIdx.x & 31;
  int gw   = blockIdx.x * 8 + wave;        // 0..2047
  int part = gw & (kSplit - 1);
  int tile = gw >> 2;                      // 0..511 : 16 l-rows each
  int b    = tile >> 7;                    // tile / (kL/16)
  int l0   = (tile & 127) << 4;
  int ml   = lane & 15;
  int kh   = (lane >> 4) << 1;             // 0 for low half, 2 for high half

  const float* ub = u + (size_t)b * kD * kL;
  v8f accB = {0.f, 0.f, 0.f, 0.f, 0.f, 0.f, 0.f, 0.f};
  v8f accC = {0.f, 0.f, 0.f, 0.f, 0.f, 0.f, 0.f, 0.f};

  int kbeg = part * kKS;
#pragma unroll 8
  for (int k = kbeg; k < kbeg + kKS; k += 4) {
    int d0 = k + kh;
    v2f A;
    A.x = ub[(size_t)d0 * kL + l0 + ml];            // (M=ml, K=kh+0)
    A.y = ub[(size_t)(d0 + 1) * kL + l0 + ml];      // (M=ml, K=kh+1)
    const float* base = WtP + (size_t)(d0 >> 1) * 64;   // 8B-aligned pairs
    v2f Bb = *(const v2f*)(base + ml * 2);          // Bx weights, K=kh,kh+1
    v2f Bc = *(const v2f*)(base + 32 + ml * 2);     // Cx weights
    accB = __builtin_amdgcn_wmma_f32_16x16x4_f32(false, A, false, Bb,
                                                 (short)0, accB, false, false);
    accC = __builtin_amdgcn_wmma_f32_16x16x4_f32(false, A, false, Bc,
                                                 (short)0, accC, false, false);
  }

  float* outp = BCp + (size_t)part * BCP_SZ;
  int mbase = (lane >> 4) << 3;   // 0 or 8
#pragma unroll
  for (int r = 0; r < 8; ++r) {
    size_t row = (size_t)(b * kL + l0 + mbase + r) * 32;
    outp[row + ml]      = accB[r];
    outp[row + 16 + ml] = accC[r];
  }
}

// ---------------------------------------------------------------------------
// 5) local scans: thread = (b, chunk, n); sums the 4 K-partials, scans 32 steps
//    in place into partial 0, and emits the chunk-final value.
__global__ void k_scan_local(float* __restrict__ BCp,
                             const float* __restrict__ A_vec,
                             float* __restrict__ carry) {
  int t = blockIdx.x * blockDim.x + threadIdx.x;   // 0..4095
  int n = t & 15;
  int c = (t >> 4) & (kNChunk - 1);
  int b = t >> 10;
  float decay = __expf(A_vec[n] * kAScale);
  float h = 0.f;
  size_t base = (size_t)(b * kL + c * kChunk) * 32 + n;
  for (int i = 0; i < kChunk; ++i) {
    size_t idx = base + (size_t)i * 32;
    float bx = BCp[idx] + BCp[idx + BCP_SZ] + BCp[idx + 2 * BCP_SZ] + BCp[idx + 3 * BCP_SZ];
    h = fmaf(decay, h, bx);
    BCp[idx] = h;          // H_local overwrites Bx partial 0
  }
  carry[(size_t)(b * kN + n) * kNChunk + c] = h;
}

// 6) carry scan across chunks: carry[c] <- full-scan value at end of chunk c-1
__global__ void k_scan_carry(float* __restrict__ carry,
                             const float* __restrict__ A_vec) {
  int t = threadIdx.x;   // 0..63
  int n = t & 15, b = t >> 4;
  float d32 = __expf(A_vec[n] * kAScale * (float)kChunk);
  float E = 0.f;
  size_t base = (size_t)(b * kN + n) * kNChunk;
  for (int c = 0; c < kNChunk; ++c) {
    float f = carry[base + c];
    carry[base + c] = E;
    E = fmaf(d32, E, f);
  }
}

// 7) combine: s[b,l] = sum_n Cx * (H_local + decay^(i+1)*carry) ; g = s/rms
__global__ void k_combine(const float* __restrict__ BCp,
                          const float* __restrict__ carry,
                          const float* __restrict__ A_vec,
                          const float* __restrict__ msq_p,
                          float* __restrict__ g) {
  int t = blockIdx.x * blockDim.x + threadIdx.x;  // 0..8191 == b*kL + l
  int b = t >> 11;
  int l = t & (kL - 1);
  int i = l & (kChunk - 1);
  int c = l >> 5;
  size_t row = (size_t)t * 32;

  float hloc[kN], cx[kN];
#pragma unroll
  for (int q = 0; q < 4; ++q) {                 // H_local: 4 x b128
    float4 v = ((const float4*)(BCp + row))[q];
    hloc[q * 4 + 0] = v.x; hloc[q * 4 + 1] = v.y;
    hloc[q * 4 + 2] = v.z; hloc[q * 4 + 3] = v.w;
  }
#pragma unroll
  for (int n = 0; n < kN; ++n) cx[n] = 0.f;
#pragma unroll
  for (int p = 0; p < kSplit; ++p) {            // Cx partial sums: 16 x b128
    const float4* c4 = (const float4*)(BCp + row + 16 + (size_t)p * BCP_SZ);
#pragma unroll
    for (int q = 0; q < 4; ++q) {
      float4 v = c4[q];
      cx[q * 4 + 0] += v.x; cx[q * 4 + 1] += v.y;
      cx[q * 4 + 2] += v.z; cx[q * 4 + 3] += v.w;
    }
  }

  float s = 0.f;
#pragma unroll
  for (int n = 0; n < kN; ++n) {
    float carr = carry[(size_t)(b * kN + n) * kNChunk + c];
    float h = fmaf(__expf(A_vec[n] * kAScale * (float)(i + 1)), carr, hloc[n]);
    s = fmaf(cx[n], h, s);
  }
  float msq = msq_p[0];
  g[t] = s * rsqrtf(fmaf(s * s, msq, kEps));
}

// ---------------------------------------------------------------------------
// 8) rank-1 output: out[b,e,l] = g[b,l]*c2[e]; one block per (b,e) row, float4
__global__ void k_outer(const float* __restrict__ g,
                        const float* __restrict__ c2,
                        float* __restrict__ out) {
  int row = blockIdx.x;          // b*kD + e
  int b = row >> 10;
  int e = row & (kD - 1);
  float cv = c2[e];
  const float4* g4 = (const float4*)(g + (size_t)b * kL);
  float4* o4 = (float4*)(out + (size_t)row * kL);
  float4 gv = g4[threadIdx.x];
  o4[threadIdx.x] = make_float4(gv.x * cv, gv.y * cv, gv.z * cv, gv.w * cv);
}

// ---------------------------------------------------------------------------
extern "C" void kernel_launch(void* const* d_in, const int* in_sizes, int n_in,
                              void* d_out, int out_size, void* d_ws, size_t ws_size,
                              hipStream_t stream) {
  (void)in_sizes; (void)n_in; (void)out_size; (void)ws_size;
  // setup_inputs order: L, u, W_in, W_B, W_C, W_delta(unused), W_out, A_vec, norm_w
  const float* u      = (const float*)d_in[1];
  const float* W_in   = (const float*)d_in[2];
  const float* W_B    = (const float*)d_in[3];
  const float* W_C    = (const float*)d_in[4];
  const float* W_out  = (const float*)d_in[6];
  const float* A_vec  = (const float*)d_in[7];
  const float* norm_w = (const float*)d_in[8];
  float* out = (float*)d_out;

  float* ws    = (float*)d_ws;
  float* WtP   = ws + WS_WT;
  float* BCp   = ws + WS_BCP;
  float* carry = ws + WS_CARRY;
  float* g     = ws + WS_G;
  float* w_sum = ws + WS_WSUM;
  float* c2    = ws + WS_C2;
  float* msq   = ws + WS_MSQ;

  k_prep_weights<<<kD / 256, 256, 0, stream>>>(W_in, W_B, W_C, WtP);
  k_wsum<<<kD / 8, 256, 0, stream>>>(W_out, norm_w, w_sum, c2);
  k_msq<<<1, 256, 0, stream>>>(w_sum, msq);
  k_gemm_bc<<<(512 * kSplit) / 8, 256, 0, stream>>>(u, WtP, BCp);
  k_scan_local<<<(kB * kN * kNChunk) / 256, 256, 0, stream>>>(BCp, A_vec, carry);
  k_scan_carry<<<1, 64, 0, stream>>>(carry, A_vec);
  k_combine<<<(kB * kL) / 256, 256, 0, stream>>>(BCp, carry, A_vec, msq, g);
  k_outer<<<kB * kD, kL / 4, 0, stream>>>(g, c2, out);
}